// CoupledStateSpace_27255862461153
// MI455X (gfx1250) — compile-verified
//
#include <hip/hip_runtime.h>

#define XDIM 64
#define BATCH 32
#define SEQ 2048
#define DTC (1.0f/1024.0f)
#define EPSC 1e-4f
#define NT 256

typedef __attribute__((ext_vector_type(2))) float v2f;
typedef __attribute__((ext_vector_type(8))) float v8f;

// ---------------- workspace layout (float offsets) ----------------
#define OFF_MEAN   0
#define OFF_STD    64
#define OFF_Q      128
#define OFF_R      (OFF_Q    + 4096)
#define OFF_VM     (OFF_R    + 4096)
#define OFF_QP     (OFF_VM   + 4096)
#define OFF_RINV0  (OFF_QP   + 4096)
#define OFF_RINV   (OFF_RINV0+ 4096)
#define OFF_H      (OFF_RINV + 4096)   // 128x128
#define OFF_GJ     (OFF_H    + 16384)  // 64x128 gauss-jordan
#define OFF_SL     (OFF_GJ   + 8192)   // 128x64 LQR riccati state
#define OFF_SNEW   (OFF_SL   + 8192)   // 128x64
#define OFF_X      (OFF_SNEW + 8192)
#define OFF_P      (OFF_X    + 4096)
#define OFF_T1     (OFF_P    + 4096)
#define OFF_K      (OFF_T1   + 4096)
#define OFF_CTP    (OFF_K    + 4096)
#define OFF_KG     (OFF_CTP  + 4096)
#define OFF_TU     (OFF_KG   + 4096)
#define OFF_TV     (OFF_TU   + 4096)
#define OFF_XP     (OFF_TV   + 4096)   // 32x64
#define OFF_RES    (OFF_XP   + 2048)   // 32x64
#define OFF_RS     (OFF_RES  + 2048)   // 32x64
#define OFF_CHOL   (OFF_RS   + 2048)
#define OFF_LOGDET (OFF_CHOL + 4096)

// ---------------- block-cooperative f32 WMMA matmul ----------------
// C[MxN] = beta*C + alpha * op(A)[MxK] * op(B)[KxN]; row-major w/ leading dims.
// tA: A stored transposed (A[k*lda+m]); tB: B stored transposed (B[n*ldb+k]).
// M,N multiples of 16; K multiple of 4. Wave32 only; uses v_wmma_f32_16x16x4_f32.
__device__ __forceinline__ void mm_wmma(const float* A, int lda, int tA,
                                        const float* B, int ldb, int tB,
                                        float* C, int ldc,
                                        int M, int N, int K,
                                        float alpha, float beta)
{
    const int lane   = threadIdx.x & 31;
    const int wave   = threadIdx.x >> 5;
    const int nwaves = blockDim.x >> 5;
    const int tN = N >> 4;
    const int ntiles = (M >> 4) * tN;
    const int lm = lane & 15;          // row (A) / col (B) within fragment
    const int lk = (lane >> 4) << 1;   // k pair base within fragment
    for (int tile = wave; tile < ntiles; tile += nwaves) {
        const int m0 = (tile / tN) << 4;
        const int n0 = (tile % tN) << 4;
        v8f acc = {0.f,0.f,0.f,0.f,0.f,0.f,0.f,0.f};
        for (int k0 = 0; k0 < K; k0 += 4) {
            const int ka = k0 + lk;
            v2f a, b;
            if (tA) { a[0] = A[ka*lda + m0+lm];     a[1] = A[(ka+1)*lda + m0+lm]; }
            else    { a[0] = A[(m0+lm)*lda + ka];   a[1] = A[(m0+lm)*lda + ka+1]; }
            if (tB) { b[0] = B[(n0+lm)*ldb + ka];   b[1] = B[(n0+lm)*ldb + ka+1]; }
            else    { b[0] = B[ka*ldb + n0+lm];     b[1] = B[(ka+1)*ldb + n0+lm]; }
            acc = __builtin_amdgcn_wmma_f32_16x16x4_f32(false, a, false, b,
                                                        (short)0, acc, false, false);
        }
        const int r0 = m0 + ((lane >> 4) << 3);
        const int c0 = n0 + lm;
#pragma unroll
        for (int i = 0; i < 8; ++i) {
            const int idx = (r0 + i) * ldc + c0;
            float prev = (beta == 0.f) ? 0.f : beta * C[idx];
            C[idx] = prev + alpha * acc[i];
        }
    }
}

// ---------------- block-cooperative 64x64 Gauss-Jordan inverse ----------------
// g: 64 x 128 augmented buffer (row stride 128); input in left half.
// On exit, inverse is in g[r*128 + 64 + c].
__device__ void gj_inverse64(float* g, float* colf)
{
    const int tid = threadIdx.x, nt = blockDim.x;
    for (int i = tid; i < 64*64; i += nt) {
        int r = i >> 6, c = i & 63;
        g[r*128 + 64 + c] = (r == c) ? 1.f : 0.f;
    }
    __syncthreads();
    for (int p = 0; p < 64; ++p) {
        for (int r = tid; r < 64; r += nt) colf[r] = g[r*128 + p];
        __syncthreads();
        const float ip = 1.f / colf[p];
        for (int c = tid; c < 128; c += nt) g[p*128 + c] *= ip;
        __syncthreads();
        for (int i = tid; i < 64*128; i += nt) {
            int r = i >> 7, c = i & 127;
            if (r != p) g[r*128 + c] -= colf[r] * g[p*128 + c];
        }
        __syncthreads();
    }
}

// ---------------- per-feature mean/std (ddof=1) ----------------
__global__ void css_stats_kernel(const float* __restrict__ y, float* __restrict__ ws)
{
    __shared__ float s1[NT], s2[NT];
    const int d = blockIdx.x;
    float a = 0.f, b = 0.f;
    for (int i = threadIdx.x; i < BATCH*SEQ; i += blockDim.x) {
        float v = y[(size_t)i*64 + d];
        a += v; b += v*v;
    }
    s1[threadIdx.x] = a; s2[threadIdx.x] = b;
    __syncthreads();
    for (int s = NT>>1; s > 0; s >>= 1) {
        if (threadIdx.x < s) { s1[threadIdx.x] += s1[threadIdx.x+s]; s2[threadIdx.x] += s2[threadIdx.x+s]; }
        __syncthreads();
    }
    if (threadIdx.x == 0) {
        const float n = (float)(BATCH*SEQ);
        const float m = s1[0] / n;
        const float var = (s2[0] - n*m*m) / (n - 1.f);
        ws[OFF_MEAN + d] = m;
        ws[OFF_STD  + d] = sqrtf(fmaxf(var, 0.f));
    }
}

__global__ void css_normalize_kernel(const float* __restrict__ y,
                                     const float* __restrict__ ws,
                                     float* __restrict__ yn)
{
    const size_t N = (size_t)BATCH*SEQ*64;
    for (size_t i = (size_t)blockIdx.x*blockDim.x + threadIdx.x; i < N;
         i += (size_t)gridDim.x*blockDim.x) {
        const int d = (int)(i & 63);
        yn[i] = (y[i] - ws[OFF_MEAN + d]) / (ws[OFF_STD + d] + EPSC);
    }
}

// ---------------- one-time precompute: Gram matrices, inverses, H_care ----------------
__global__ void css_setup_kernel(const float* __restrict__ Amat, const float* __restrict__ Bmat,
                                 const float* __restrict__ LQ,  const float* __restrict__ LR,
                                 const float* __restrict__ LV,  const float* __restrict__ LQp,
                                 float* __restrict__ ws)
{
    __shared__ float gj[64*128];
    __shared__ float colf[64];
    const int tid = threadIdx.x, nt = blockDim.x;
    float* Q     = ws + OFF_Q;
    float* R     = ws + OFF_R;
    float* Vm    = ws + OFF_VM;
    float* Qp    = ws + OFF_QP;
    float* Rinv0 = ws + OFF_RINV0;
    float* Rinv  = ws + OFF_RINV;
    float* H     = ws + OFF_H;
    float* T1    = ws + OFF_T1;
    float* T2    = ws + OFF_P;

    mm_wmma(LQ,64,0,  LQ,64,1,  Q,64,  64,64,64, 1.f,0.f); __syncthreads();
    mm_wmma(LR,64,0,  LR,64,1,  R,64,  64,64,64, 1.f,0.f); __syncthreads();
    mm_wmma(LV,64,0,  LV,64,1,  Vm,64, 64,64,64, 1.f,0.f); __syncthreads();
    mm_wmma(LQp,64,0, LQp,64,1, Qp,64, 64,64,64, 1.f,0.f); __syncthreads();

    // Rinv0 = inv(R)
    for (int i = tid; i < 4096; i += nt) { int r=i>>6,c=i&63; gj[r*128+c] = R[i]; }
    __syncthreads();
    gj_inverse64(gj, colf);
    for (int i = tid; i < 4096; i += nt) { int r=i>>6,c=i&63; Rinv0[i] = gj[r*128+64+c]; }
    __syncthreads();
    // Rinv = inv(R + eps I)
    for (int i = tid; i < 4096; i += nt) { int r=i>>6,c=i&63; gj[r*128+c] = R[i] + ((r==c)?EPSC:0.f); }
    __syncthreads();
    gj_inverse64(gj, colf);
    for (int i = tid; i < 4096; i += nt) { int r=i>>6,c=i&63; Rinv[i] = gj[r*128+64+c]; }
    __syncthreads();

    // H_care = [[A, -B inv(R) B^T], [-Q, -A^T]]
    mm_wmma(Bmat,64,0, Rinv0,64,0, T1,64, 64,64,64, 1.f,0.f); __syncthreads();
    mm_wmma(T1,64,0,   Bmat,64,1,  T2,64, 64,64,64, 1.f,0.f); __syncthreads();
    for (int i = tid; i < 4096; i += nt) {
        int r = i>>6, c = i&63;
        H[r*128 + c]            =  Amat[i];
        H[r*128 + 64 + c]       = -T2[i];
        H[(64+r)*128 + c]       = -Q[i];
        H[(64+r)*128 + 64 + c]  = -Amat[c*64 + r];
    }
    __syncthreads();

    // log|V + eps I| via Cholesky (runs once; serial on one lane)
    if (tid == 0) {
        float* ch = ws + OFF_CHOL;
        for (int i = 0; i < 4096; ++i) ch[i] = Vm[i];
        for (int i = 0; i < 64; ++i)   ch[i*64+i] += EPSC;
        float ld = 0.f;
        for (int k = 0; k < 64; ++k) {
            float d = ch[k*64+k];
            for (int j = 0; j < k; ++j) { float v = ch[k*64+j]; d -= v*v; }
            d = sqrtf(fmaxf(d, 1e-30f));
            ch[k*64+k] = d; ld += __logf(d);
            const float idd = 1.f / d;
            for (int r = k+1; r < 64; ++r) {
                float s = ch[r*64+k];
                for (int j = 0; j < k; ++j) s -= ch[r*64+j]*ch[k*64+j];
                ch[r*64+k] = s * idd;
            }
        }
        ws[OFF_LOGDET] = 2.f * ld;
    }
}

// ---------------- fused sequential scan: LQR + KF + outputs ----------------
__global__ void css_fused_kernel(const float* __restrict__ Amat, const float* __restrict__ Bmat,
                                 const float* __restrict__ Cmat,
                                 const float* __restrict__ x0_mean, const float* __restrict__ P0,
                                 const float* __restrict__ yn,
                                 float* __restrict__ ws, float* __restrict__ out)
{
    __shared__ float Skf[128*64];   // KF Riccati state
    __shared__ float xs[BATCH*64];  // filtered states
    __shared__ float Aeff[64*64];
    __shared__ float colf[64];
    __shared__ float red[NT];
    __shared__ float xc[BATCH], uc[BATCH];
    const int tid = threadIdx.x, nt = blockDim.x;

    float* H    = ws + OFF_H;
    float* GJ   = ws + OFF_GJ;
    float* SL   = ws + OFF_SL;
    float* Snew = ws + OFF_SNEW;
    float* X    = ws + OFF_X;
    float* P    = ws + OFF_P;
    float* T1   = ws + OFF_T1;
    float* K    = ws + OFF_K;
    float* CtP  = ws + OFF_CTP;
    float* Kg   = ws + OFF_KG;
    float* TU   = ws + OFF_TU;
    float* TV   = ws + OFF_TV;
    float* xp   = ws + OFF_XP;
    float* resid= ws + OFF_RES;
    float* rs   = ws + OFF_RS;
    float* Q    = ws + OFF_Q;
    float* R    = ws + OFF_R;
    float* Vm   = ws + OFF_VM;
    float* Qp   = ws + OFF_QP;
    float* Rinv = ws + OFF_RINV;
    const float logdetV = ws[OFF_LOGDET];
    const float LOG2PI  = 1.8378770664093453f;

    float* xout  = out + 2;
    float* uout  = out + 2 + (size_t)BATCH*SEQ*64;
    float* ypout = out + 2 + 2*(size_t)BATCH*SEQ*64;

    // init states
    for (int i = tid; i < 128*64; i += nt) {
        int r = i >> 6, c = i & 63;
        SL[i]  = (r < 64) ? ((r==c)?1.f:0.f) : Q[(r-64)*64 + c];
        Skf[i] = (r < 64) ? ((r==c)?1.f:0.f) : P0[(r-64)*64 + c];
    }
    for (int i = tid; i < BATCH*64; i += nt) xs[i] = x0_mean[i & 63];
    if (tid < BATCH) { xc[tid] = 0.f; uc[tid] = 0.f; }
    __syncthreads();

    float ll = 0.f;

    for (int t = 0; t < SEQ; ++t) {
        // ===== LQR Riccati step: S += DT*(H_care @ S) =====
        mm_wmma(H,128,0, SL,64,0, Snew,64, 128,64,128, DTC, 0.f); __syncthreads();
        for (int i = tid; i < 128*64; i += nt) SL[i] += Snew[i];
        __syncthreads();
        // X = V @ inv(U)
        for (int i = tid; i < 4096; i += nt) { int r=i>>6,c=i&63; GJ[r*128+c] = SL[i]; }
        __syncthreads();
        gj_inverse64(GJ, colf);
        mm_wmma(SL+64*64,64,0, GJ+64,128,0, X,64, 64,64,64, 1.f,0.f); __syncthreads();
        for (int i = tid; i < 4096; i += nt) { int r=i>>6,c=i&63; P[i] = 0.5f*(X[i] + X[c*64+r]); }
        __syncthreads();
        // K_t = Rinv_reg @ (B^T P)
        mm_wmma(Bmat,64,1, P,64,0,  T1,64, 64,64,64, 1.f,0.f); __syncthreads();
        mm_wmma(Rinv,64,0, T1,64,0, K,64,  64,64,64, 1.f,0.f); __syncthreads();

        // ===== KF step =====
        // A_eff = A - B K_t
        for (int i = tid; i < 4096; i += nt) Aeff[i] = Amat[i];
        __syncthreads();
        mm_wmma(Bmat,64,0, K,64,0, Aeff,64, 64,64,64, -1.f,1.f); __syncthreads();
        // x_pred = x + DT * x A_eff^T
        for (int i = tid; i < BATCH*64; i += nt) xp[i] = xs[i];
        __syncthreads();
        mm_wmma(xs,64,0, Aeff,64,1, xp,64, BATCH,64,64, DTC,1.f); __syncthreads();
        // Riccati (block-structured H_t): U += DT*(Aeff U + Qp V); V -= DT*(Aeff^T V)
        mm_wmma(Aeff,64,0, Skf,64,0,       TU,64, 64,64,64, 1.f,0.f); __syncthreads();
        mm_wmma(Qp,64,0,   Skf+64*64,64,0, TU,64, 64,64,64, 1.f,1.f); __syncthreads();
        mm_wmma(Aeff,64,1, Skf+64*64,64,0, TV,64, 64,64,64, 1.f,0.f); __syncthreads();
        for (int i = tid; i < 4096; i += nt) {
            Skf[i]       += DTC * TU[i];
            Skf[64*64+i] -= DTC * TV[i];
        }
        __syncthreads();
        // P_pred = sym(V @ inv(U))
        for (int i = tid; i < 4096; i += nt) { int r=i>>6,c=i&63; GJ[r*128+c] = Skf[i]; }
        __syncthreads();
        gj_inverse64(GJ, colf);
        mm_wmma(Skf+64*64,64,0, GJ+64,128,0, X,64, 64,64,64, 1.f,0.f); __syncthreads();
        for (int i = tid; i < 4096; i += nt) { int r=i>>6,c=i&63; P[i] = 0.5f*(X[i] + X[c*64+r]); }
        __syncthreads();
        // residual = y_t - x_pred C^T
        for (int i = tid; i < BATCH*64; i += nt) {
            int b = i >> 6, d = i & 63;
            resid[i] = yn[(size_t)b*SEQ*64 + (size_t)t*64 + d];
        }
        __syncthreads();
        mm_wmma(xp,64,0, Cmat,64,1, resid,64, BATCH,64,64, -1.f,1.f); __syncthreads();
        // innovation covariance + inverse
        mm_wmma(Cmat,64,0, P,64,0,   CtP,64, 64,64,64, 1.f,0.f); __syncthreads();
        mm_wmma(CtP,64,0,  Cmat,64,1, GJ,128, 64,64,64, 1.f,0.f); __syncthreads();
        for (int i = tid; i < 4096; i += nt) { int r=i>>6,c=i&63; GJ[r*128+c] += Vm[i] + ((r==c)?EPSC:0.f); }
        __syncthreads();
        gj_inverse64(GJ, colf);
        // Kg = CtP^T @ S_inv
        mm_wmma(CtP,64,1, GJ+64,128,0, Kg,64, 64,64,64, 1.f,0.f); __syncthreads();
        // x_new = x_pred + resid Kg^T
        for (int i = tid; i < BATCH*64; i += nt) xs[i] = xp[i];
        __syncthreads();
        mm_wmma(resid,64,0, Kg,64,1, xs,64, BATCH,64,64, 1.f,1.f); __syncthreads();
        // quad = sum(resid * (resid S_inv^T)); ll update
        mm_wmma(resid,64,0, GJ+64,128,1, rs,64, BATCH,64,64, 1.f,0.f); __syncthreads();
        float part = 0.f;
        for (int i = tid; i < BATCH*64; i += nt) part += resid[i]*rs[i];
        red[tid] = part; __syncthreads();
        for (int s = nt>>1; s > 0; s >>= 1) { if (tid < s) red[tid] += red[tid+s]; __syncthreads(); }
        ll -= 0.5f * (red[0] + logdetV + 64.f*LOG2PI);
        __syncthreads();

        // ===== outputs for step t =====
        for (int i = tid; i < BATCH*64; i += nt) {
            int b = i >> 6, d = i & 63;
            xout[(size_t)b*SEQ*64 + (size_t)t*64 + d] = xs[i];
        }
        // u = -x K_t^T  (reuse xp as u)
        mm_wmma(xs,64,0, K,64,1, xp,64, BATCH,64,64, -1.f,0.f); __syncthreads();
        for (int i = tid; i < BATCH*64; i += nt) {
            int b = i >> 6, d = i & 63;
            uout[(size_t)b*SEQ*64 + (size_t)t*64 + d] = xp[i];
        }
        // y_pred = x C^T (reuse rs)
        mm_wmma(xs,64,0, Cmat,64,1, rs,64, BATCH,64,64, 1.f,0.f); __syncthreads();
        for (int i = tid; i < BATCH*64; i += nt) {
            int b = i >> 6, d = i & 63;
            ypout[(size_t)b*SEQ*64 + (size_t)t*64 + d] = rs[i];
        }
        // costs: x Q x and u R u (reuse resid)
        mm_wmma(xs,64,0, Q,64,0, resid,64, BATCH,64,64, 1.f,0.f); __syncthreads();
        if (tid < BATCH) {
            float s = 0.f;
            for (int j = 0; j < 64; ++j) s += xs[tid*64+j]*resid[tid*64+j];
            xc[tid] += s;
        }
        __syncthreads();
        mm_wmma(xp,64,0, R,64,0, resid,64, BATCH,64,64, 1.f,0.f); __syncthreads();
        if (tid < BATCH) {
            float s = 0.f;
            for (int j = 0; j < 64; ++j) s += xp[tid*64+j]*resid[tid*64+j];
            uc[tid] += s;
        }
        __syncthreads();
    }

    if (tid == 0) {
        out[0] = ll / (float)SEQ;
        float s = 0.f;
        for (int b = 0; b < BATCH; ++b) s += xc[b] + uc[b];
        out[1] = s / (float)BATCH;
    }
}

extern "C" void kernel_launch(void* const* d_in, const int* in_sizes, int n_in,
                              void* d_out, int out_size, void* d_ws, size_t ws_size,
                              hipStream_t stream)
{
    const float* y   = (const float*)d_in[0];
    const float* A   = (const float*)d_in[1];
    const float* B   = (const float*)d_in[2];
    const float* C   = (const float*)d_in[3];
    const float* LQ  = (const float*)d_in[4];
    const float* LR  = (const float*)d_in[5];
    const float* LV  = (const float*)d_in[6];
    const float* LQp = (const float*)d_in[7];
    const float* x0  = (const float*)d_in[8];
    const float* P0  = (const float*)d_in[9];

    float* out = (float*)d_out;
    float* ws  = (float*)d_ws;
    float* yn  = out + 2 + 3*(size_t)BATCH*SEQ*64;   // yn output region, also KF input

    css_stats_kernel    <<<64,   NT, 0, stream>>>(y, ws);
    css_normalize_kernel<<<2048, NT, 0, stream>>>(y, ws, yn);
    css_setup_kernel    <<<1,    NT, 0, stream>>>(A, B, LQ, LR, LV, LQp, ws);
    css_fused_kernel    <<<1,    NT, 0, stream>>>(A, B, C, x0, P0, yn, ws, out);
}